// TripletLossL1_13967233646987
// MI455X (gfx1250) — compile-verified
//
#include <hip/hip_runtime.h>
#include <hip/hip_bf16.h>
#include <stdint.h>

#define N_SAMPLES 512
#define DIM       256
#define BI        32
#define BJ        32
#define CHUNK     128            // dims staged per LDS pass (2 passes)
#define LDW       (CHUNK + 4)    // padded LDS row stride in floats (16B aligned, bank-skewed)
#define MARGIN_F  0.3f

// ---- CDNA5 async global->LDS path (guarded; falls back to plain copy) ----
#if __has_builtin(__builtin_amdgcn_global_load_async_to_lds_b128)
#define HAVE_ASYNC_B128 1
#else
#define HAVE_ASYNC_B128 0
#endif

typedef int v4i_t __attribute__((ext_vector_type(4)));
typedef __attribute__((address_space(1))) v4i_t* g_v4i_ptr;   // global
typedef __attribute__((address_space(3))) v4i_t* l_v4i_ptr;   // LDS

__device__ __forceinline__ void copy16_to_lds(const float* g, float* l) {
#if HAVE_ASYNC_B128
    __builtin_amdgcn_global_load_async_to_lds_b128(
        (g_v4i_ptr)g, (l_v4i_ptr)l, /*imm offset*/ 0, /*cpol*/ 0);
#else
    *(float4*)l = *(const float4*)g;
#endif
}

__device__ __forceinline__ void wait_async_lds() {
#if HAVE_ASYNC_B128
#if __has_builtin(__builtin_amdgcn_s_wait_asynccnt)
    __builtin_amdgcn_s_wait_asynccnt(0);
#else
    asm volatile("s_wait_asynccnt 0" ::: "memory");
#endif
#endif
}

// ---------------- kernel 0: init workspace (ap=0 max identity, an=+inf) ---
__global__ void tl_init_kernel(unsigned* apb, unsigned* anb) {
    int i = threadIdx.x;
    if (i < N_SAMPLES) {
        apb[i] = 0u;            // float 0.0 ; ap >= 0 always (diagonal is same-class, dist 0)
        anb[i] = 0x7f800000u;   // +inf
    }
}

// ---------------- kernel 1: 32x32 distance tiles + hard mining ------------
__global__ __launch_bounds__(256) void tl_tiles_kernel(
    const float* __restrict__ x, const int* __restrict__ tgt,
    unsigned* __restrict__ apb, unsigned* __restrict__ anb) {

    __shared__ __align__(16) float sXi[BI * LDW];
    __shared__ __align__(16) float sXj[BJ * LDW];

    const int tid  = threadIdx.x;
    const int i0   = blockIdx.x * BI;
    const int j0   = blockIdx.y * BJ;
    const int jj   = tid & 31;    // j within tile == lane
    const int ig   = tid >> 5;    // wave id: i-group (4 rows each)
    const int lane = tid & 31;

    float acc[4] = {0.f, 0.f, 0.f, 0.f};

    for (int h = 0; h < 2; ++h) {
        // stage chunk h of both tiles: 2 tiles * 32 rows * 32 float4 = 2048 xfers
        #pragma unroll
        for (int k = 0; k < 8; ++k) {
            int t    = tid + k * 256;
            int tile = t >> 10;          // 0: i-tile, 1: j-tile
            int rem  = t & 1023;
            int row  = rem >> 5;
            int qd   = rem & 31;         // float4 index within chunk
            int grow = tile ? (j0 + row) : (i0 + row);
            const float* src = x + (size_t)grow * DIM + h * CHUNK + qd * 4;
            float*       dst = (tile ? sXj : sXi) + row * LDW + qd * 4;
            copy16_to_lds(src, dst);
        }
        wait_async_lds();        // this wave's async transfers landed in LDS
        __syncthreads();         // all waves' data visible

        #pragma unroll 4
        for (int q = 0; q < CHUNK / 4; ++q) {
            float4 xj = *(const float4*)&sXj[jj * LDW + q * 4];
            #pragma unroll
            for (int r = 0; r < 4; ++r) {
                float4 xi = *(const float4*)&sXi[(ig * 4 + r) * LDW + q * 4];
                acc[r] += fabsf(xi.x - xj.x) + fabsf(xi.y - xj.y) +
                          fabsf(xi.z - xj.z) + fabsf(xi.w - xj.w);
            }
        }
        __syncthreads();         // done reading before next chunk overwrites
    }

    // hard mining within the tile: reduce over the 32 j-lanes of this wave
    const int tj = tgt[j0 + jj];
    const float INF = __uint_as_float(0x7f800000u);

    #pragma unroll
    for (int r = 0; r < 4; ++r) {
        int   i    = i0 + ig * 4 + r;
        int   ti   = tgt[i];
        float dist = acc[r] * (1.0f / (float)DIM);
        float apv  = (ti == tj) ? dist : 0.0f;  // 0 filler is harmless: ap >= 0
        float anv  = (ti == tj) ? INF  : dist;  // +inf filler is harmless for min
        #pragma unroll
        for (int off = 16; off > 0; off >>= 1) {
            apv = fmaxf(apv, __shfl_xor(apv, off, 32));
            anv = fminf(anv, __shfl_xor(anv, off, 32));
        }
        if (lane == 0) {
            // non-negative floats: uint bit order == value order
            atomicMax(&apb[i], __float_as_uint(apv));
            atomicMin(&anb[i], __float_as_uint(anv));
        }
    }
}

// ---------------- kernel 2: loss = mean(relu(ap - an + margin)) -----------
__global__ void tl_finalize_kernel(const unsigned* __restrict__ apb,
                                   const unsigned* __restrict__ anb,
                                   float* __restrict__ out) {
    __shared__ float red[16];
    const int i    = threadIdx.x;   // 512 threads = 16 waves
    const int lane = i & 31;
    const int wid  = i >> 5;

    float ap = __uint_as_float(apb[i]);
    float an = __uint_as_float(anb[i]);
    float l  = fmaxf(ap - an + MARGIN_F, 0.0f);

    #pragma unroll
    for (int off = 16; off > 0; off >>= 1) l += __shfl_xor(l, off, 32);
    if (lane == 0) red[wid] = l;
    __syncthreads();

    if (i < 32) {                  // whole wave 0 active for the shuffles
        float v = (i < 16) ? red[i] : 0.0f;
        #pragma unroll
        for (int off = 8; off > 0; off >>= 1) v += __shfl_xor(v, off, 32);
        if (i == 0) out[0] = v * (1.0f / (float)N_SAMPLES);
    }
}

extern "C" void kernel_launch(void* const* d_in, const int* in_sizes, int n_in,
                              void* d_out, int out_size, void* d_ws, size_t ws_size,
                              hipStream_t stream) {
    (void)in_sizes; (void)n_in; (void)out_size; (void)ws_size;
    const float* x   = (const float*)d_in[0];
    const int*   tgt = (const int*)d_in[1];
    float*       out = (float*)d_out;

    unsigned* apb = (unsigned*)d_ws;            // 512 u32
    unsigned* anb = apb + N_SAMPLES;            // 512 u32

    tl_init_kernel<<<1, N_SAMPLES, 0, stream>>>(apb, anb);
    dim3 grid(N_SAMPLES / BI, N_SAMPLES / BJ);  // 16 x 16 = 256 blocks
    tl_tiles_kernel<<<grid, 256, 0, stream>>>(x, tgt, apb, anb);
    tl_finalize_kernel<<<1, N_SAMPLES, 0, stream>>>(apb, anb, out);
}